// CuGraphRelGraphConv_29326036697258
// MI455X (gfx1250) — compile-verified
//
#include <hip/hip_runtime.h>

namespace {
constexpr int kNodes   = 100000;
constexpr int kEdges   = 1600000;
constexpr int kInFeat  = 64;
constexpr int kOutFeat = 64;
constexpr int kBases   = 2;
constexpr int kAggW    = kBases * kInFeat;   // 128 (agg row width)
constexpr int kKTot    = kAggW + kInFeat;    // 192 (fused K dimension)
constexpr int kPairs   = kKTot / 2;          // 96 row-pairs of the fused weight matrix
constexpr int kBStr    = 80;                 // float2 stride per pair-row: 160 dwords == 32 mod 64
}

typedef __attribute__((ext_vector_type(2))) float v2f;
typedef __attribute__((ext_vector_type(8))) float v8f;

// ---------------------------------------------------------------- zero agg
__global__ __launch_bounds__(256) void rgcn_zero(float4* __restrict__ agg4, int n4) {
  int i = blockIdx.x * blockDim.x + threadIdx.x;
  if (i < n4) agg4[i] = make_float4(0.f, 0.f, 0.f, 0.f);
}

// ---------------------------------------------------------------- edge scatter
// 16 lanes per edge, float4 feature chunk, 8 native fp32 atomics per lane.
__global__ __launch_bounds__(256) void rgcn_scatter(
    const float* __restrict__ feat, const float* __restrict__ coeff,
    const int* __restrict__ src, const int* __restrict__ dst,
    const int* __restrict__ etypes, float* __restrict__ agg) {
  int gid = blockIdx.x * blockDim.x + threadIdx.x;
  int e = gid >> 4;
  if (e >= kEdges) return;
  int d = (gid & 15) << 2;
  int s = src[e];
  int t = dst[e];
  int r = etypes[e];
  float c0 = coeff[r * kBases + 0];
  float c1 = coeff[r * kBases + 1];
  float4 f = *(const float4*)(feat + (size_t)s * kInFeat + d);
  float* a = agg + (size_t)t * kAggW + d;
  unsafeAtomicAdd(a + 0, c0 * f.x);
  unsafeAtomicAdd(a + 1, c0 * f.y);
  unsafeAtomicAdd(a + 2, c0 * f.z);
  unsafeAtomicAdd(a + 3, c0 * f.w);
  unsafeAtomicAdd(a + kInFeat + 0, c1 * f.x);
  unsafeAtomicAdd(a + kInFeat + 1, c1 * f.y);
  unsafeAtomicAdd(a + kInFeat + 2, c1 * f.z);
  unsafeAtomicAdd(a + kInFeat + 3, c1 * f.w);
}

// ---------------------------------------------------------------- fused WMMA GEMM
// out[n,:] = [agg[n,0:128] | feat[n,0:64]] @ [W(128x64) ; loop_weight(64x64)] + bias
// Weight matrix staged in LDS as row-pair interleaved float2 so each B fragment
// is a single ds_load_b64. 8 waves/block, each wave -> 32 rows x 64 cols
// (two A fragments share every B fragment).
__global__ __launch_bounds__(256) void rgcn_gemm(
    const float* __restrict__ agg, const float* __restrict__ feat,
    const float* __restrict__ W, const float* __restrict__ loopw,
    const float* __restrict__ bias, float* __restrict__ out) {
  __shared__ float2 ldsB[kPairs * kBStr];   // 96*80*8 = 61440 B

  // Stage: ldsB[p][n] = { Wcat[2p][n], Wcat[2p+1][n] }
  for (int idx = threadIdx.x; idx < kPairs * (kOutFeat / 4); idx += blockDim.x) {
    int p  = idx >> 4;
    int c  = (idx & 15) << 2;
    int k0 = 2 * p;
    const float* s0;
    const float* s1;
    if (k0 < kAggW) {
      s0 = W + (size_t)k0 * kOutFeat;
      s1 = W + (size_t)(k0 + 1) * kOutFeat;
    } else {
      s0 = loopw + (size_t)(k0 - kAggW) * kOutFeat;
      s1 = loopw + (size_t)(k0 + 1 - kAggW) * kOutFeat;
    }
    float4 r0 = *(const float4*)(s0 + c);
    float4 r1 = *(const float4*)(s1 + c);
    float2* dp = &ldsB[p * kBStr + c];
    dp[0] = make_float2(r0.x, r1.x);
    dp[1] = make_float2(r0.y, r1.y);
    dp[2] = make_float2(r0.z, r1.z);
    dp[3] = make_float2(r0.w, r1.w);
  }
  __syncthreads();

  const int wave = threadIdx.x >> 5;
  const int lane = threadIdx.x & 31;
  const long row0 = ((long)blockIdx.x * 8 + wave) * 32;
  if (row0 >= kNodes) return;   // wave-uniform exit; live waves keep EXEC all-ones

  const int half = lane >> 4;        // 0: K+{0,1}, 1: K+{2,3}
  const int l15  = lane & 15;        // M (A-frag) / N (B-frag) within tile
  const int koff = half * 2;

  const float* x0 = agg  + (row0 + l15)      * (long)kAggW;
  const float* x1 = agg  + (row0 + 16 + l15) * (long)kAggW;
  const float* f0 = feat + (row0 + l15)      * (long)kInFeat;
  const float* f1 = feat + (row0 + 16 + l15) * (long)kInFeat;

  v8f acc[2][4] = {};

  // Phase 1: K = 0..127 over agg rows vs W part of LDS
#pragma unroll
  for (int k = 0; k < kAggW; k += 4) {
    v2f a0 = *(const v2f*)(x0 + k + koff);
    v2f a1 = *(const v2f*)(x1 + k + koff);
    const float2* bp = &ldsB[((k >> 1) + half) * kBStr + l15];
#pragma unroll
    for (int t = 0; t < 4; ++t) {
      v2f b = *(const v2f*)(bp + t * 16);
      acc[0][t] = __builtin_amdgcn_wmma_f32_16x16x4_f32(false, a0, false, b,
                                                        (short)0, acc[0][t], false, false);
      acc[1][t] = __builtin_amdgcn_wmma_f32_16x16x4_f32(false, a1, false, b,
                                                        (short)0, acc[1][t], false, false);
    }
  }

  // Phase 2: K = 128..191 over feat rows vs loop_weight part of LDS
#pragma unroll
  for (int k = 0; k < kInFeat; k += 4) {
    v2f a0 = *(const v2f*)(f0 + k + koff);
    v2f a1 = *(const v2f*)(f1 + k + koff);
    const float2* bp = &ldsB[(((kAggW + k) >> 1) + half) * kBStr + l15];
#pragma unroll
    for (int t = 0; t < 4; ++t) {
      v2f b = *(const v2f*)(bp + t * 16);
      acc[0][t] = __builtin_amdgcn_wmma_f32_16x16x4_f32(false, a0, false, b,
                                                        (short)0, acc[0][t], false, false);
      acc[1][t] = __builtin_amdgcn_wmma_f32_16x16x4_f32(false, a1, false, b,
                                                        (short)0, acc[1][t], false, false);
    }
  }

  // Store D: VGPR i -> M = i (lanes 0-15) or M = 8+i (lanes 16-31), N = l15 + 16*t
  float b0 = bias[l15 +  0];
  float b1 = bias[l15 + 16];
  float b2 = bias[l15 + 32];
  float b3 = bias[l15 + 48];
#pragma unroll
  for (int rb = 0; rb < 2; ++rb) {
    float* orow = out + (row0 + rb * 16 + half * 8) * (long)kOutFeat;
#pragma unroll
    for (int i = 0; i < 8; ++i) {
      long o = (long)i * kOutFeat + l15;
      orow[o +  0] = acc[rb][0][i] + b0;
      orow[o + 16] = acc[rb][1][i] + b1;
      orow[o + 32] = acc[rb][2][i] + b2;
      orow[o + 48] = acc[rb][3][i] + b3;
    }
  }
}

extern "C" void kernel_launch(void* const* d_in, const int* in_sizes, int n_in,
                              void* d_out, int out_size, void* d_ws, size_t ws_size,
                              hipStream_t stream) {
  const float* feat  = (const float*)d_in[0];
  const float* W     = (const float*)d_in[1];
  const float* coeff = (const float*)d_in[2];
  const float* bias  = (const float*)d_in[3];
  const float* loopw = (const float*)d_in[4];
  const int*   src   = (const int*)d_in[5];
  const int*   dst   = (const int*)d_in[6];
  const int*   et    = (const int*)d_in[7];
  float* out = (float*)d_out;
  float* agg = (float*)d_ws;   // [kNodes x 128] fp32 = 51.2 MB scratch

  // 1) zero the aggregation buffer (harness poisons d_ws)
  int n4 = kNodes * kAggW / 4;
  rgcn_zero<<<(n4 + 255) / 256, 256, 0, stream>>>((float4*)agg, n4);

  // 2) edge scatter with native fp32 atomics (L2-resident)
  long sthreads = (long)kEdges * 16;
  rgcn_scatter<<<(unsigned)((sthreads + 255) / 256), 256, 0, stream>>>(
      feat, coeff, src, dst, et, agg);

  // 3) fused WMMA GEMM: 8 waves/block * 32 rows = 256 rows per block
  int tiles32 = kNodes / 32;                 // 3125 exact
  int blocks  = (tiles32 + 7) / 8;           // 391
  rgcn_gemm<<<blocks, 256, 0, stream>>>(agg, feat, W, loopw, bias, out);
}